// rhythmLoss_36498632082235
// MI455X (gfx1250) — compile-verified
//
#include <hip/hip_runtime.h>
#include <hip/hip_bf16.h>
#include <stdint.h>

#define BB 32
#define VV 518
#define SS 2048
#define NCOL (BB * SS)  // 65536 (b,s) columns

typedef __attribute__((ext_vector_type(2))) float v2f;
typedef __attribute__((ext_vector_type(8))) float v8f;

// Exact 32-lane sum via V_WMMA_F32_16X16X4_F32.
// A = {x, 0} per lane (each lane's value appears exactly once in the 16x4 A
// matrix), B = all-ones 4x16  =>  D[m,n] = rowsum(A)[m] for every n.
// Per the CDNA5 C/D layout, lane L's 8 D-VGPRs hold rows {0..7} (L<16) or
// {8..15} (L>=16); summing them and adding the xor-16 partner yields
// sum over all A entries == sum of x over all 32 lanes, in every lane.
__device__ __forceinline__ float wave_sum_wmma(float x) {
  v2f a; a[0] = x;    a[1] = 0.0f;
  v2f b; b[0] = 1.0f; b[1] = 1.0f;
  v8f c = {};
  c = __builtin_amdgcn_wmma_f32_16x16x4_f32(false, a, false, b,
                                            (short)0, c, false, false);
  float s = ((c[0] + c[1]) + (c[2] + c[3])) + ((c[4] + c[5]) + (c[6] + c[7]));
  s += __shfl_xor(s, 16, 32);
  return s;
}

__device__ __forceinline__ int tok_type(int id) {
  return (id < 128) ? 0 : (id < 289) ? 1 : (id < 390) ? 2 : 3;
}

__global__ void zero_acc_k(float* acc) {
  acc[0] = 0.0f;
  acc[1] = 0.0f;
}

// Pass 1: per (b,v) row, logsumexp over S=2048 (contiguous).  One block per
// row; 8 floats/thread stay in registers between the max and exp-sum phases.
__global__ __launch_bounds__(256) void pass1_lse_k(const float* __restrict__ X,
                                                   float* __restrict__ lse) {
  const int row = blockIdx.x;                    // 0 .. B*V-1
  const int tid = threadIdx.x;
  const float4* rp4 = (const float4*)(X + (size_t)row * SS);
  float4 a = rp4[tid];
  float4 b = rp4[tid + 256];

  float m = fmaxf(fmaxf(fmaxf(a.x, a.y), fmaxf(a.z, a.w)),
                  fmaxf(fmaxf(b.x, b.y), fmaxf(b.z, b.w)));
  #pragma unroll
  for (int off = 16; off >= 1; off >>= 1)
    m = fmaxf(m, __shfl_xor(m, off, 32));

  __shared__ float smax[8];
  __shared__ float ssum[8];
  const int wid = tid >> 5, lane = tid & 31;
  if (lane == 0) smax[wid] = m;
  __syncthreads();
  float M = smax[0];
  #pragma unroll
  for (int i = 1; i < 8; ++i) M = fmaxf(M, smax[i]);

  float s = __expf(a.x - M) + __expf(a.y - M) + __expf(a.z - M) + __expf(a.w - M)
          + __expf(b.x - M) + __expf(b.y - M) + __expf(b.z - M) + __expf(b.w - M);
  float wsum = wave_sum_wmma(s);
  if (lane == 0) ssum[wid] = wsum;
  __syncthreads();
  if (tid == 0) {
    float t = ((ssum[0] + ssum[1]) + (ssum[2] + ssum[3]))
            + ((ssum[4] + ssum[5]) + (ssum[6] + ssum[7]));
    lse[row] = M + __logf(t);
  }
}

// Pass 2: per (b,s) column — online softmax over V for CE, argmax of
// (x - lse_row[v]) for pred, token-distance mask, block-reduced into acc.
__global__ __launch_bounds__(256) void pass2_ce_k(const float* __restrict__ X,
                                                  const int* __restrict__ target,
                                                  const float* __restrict__ lse,
                                                  float* __restrict__ acc) {
  const int b     = blockIdx.x >> 3;
  const int chunk = blockIdx.x & 7;
  const int tid   = threadIdx.x;
  const int s     = (chunk << 8) + tid;

  __shared__ float s_lse[VV];
  __shared__ float rn[8];
  __shared__ float rm[8];

  // CDNA5 async copy: stage this b's LSE row (518 floats) into LDS.
  {
    const float* gsrc = lse + b * VV;            // uniform -> SGPR pair
    for (int i = tid; i < VV; i += 256) {
      unsigned lds_off = (unsigned)(size_t)(&s_lse[i]);
      unsigned goff    = (unsigned)(i * 4);
      asm volatile("global_load_async_to_lds_b32 %0, %1, %2"
                   :: "v"(lds_off), "v"(goff), "s"(gsrc)
                   : "memory");
    }
    asm volatile("s_wait_asynccnt 0" ::: "memory");
  }
  __syncthreads();

  const int tgt = target[b * SS + s];
  const float* col = X + (size_t)b * VV * SS + s;

  float m = -INFINITY, sum = 0.0f, best = -INFINITY, xt = 0.0f;
  int pred = 0;
  #pragma unroll 2
  for (int v = 0; v < VV; ++v) {
    float xv = col[(size_t)v * SS];              // coalesced: lanes span s
    float nm = fmaxf(m, xv);                     // branchless online softmax
    sum = sum * __expf(m - nm) + __expf(xv - nm);
    m = nm;
    float sc = xv - s_lse[v];
    pred = (sc > best) ? v : pred;
    best = fmaxf(best, sc);
    xt = (v == tgt) ? xv : xt;
  }
  float nll = (m + __logf(sum)) - xt;            // -log_softmax at target

  const int pt = tok_type(pred), tt = tok_type(tgt);
  float mask;
  if (pt != tt) {
    mask = 1.0f;                                 // COEF[0]
  } else if (pt == 0) {
    mask = 0.5f;                                 // pitch branch: constant
  } else {
    const float dn = (pt == 1) ? 160.0f : (pt == 2) ? 100.0f : 128.0f;
    mask = 0.5f * fabsf((float)(pred - tgt)) / dn;
  }

  // Block reduction: WMMA wave-sum, then 8 wave totals via LDS, one atomic.
  float wn = wave_sum_wmma(nll);
  float wm = wave_sum_wmma(mask);
  const int wid = tid >> 5, lane = tid & 31;
  if (lane == 0) { rn[wid] = wn; rm[wid] = wm; }
  __syncthreads();
  if (tid == 0) {
    float an = ((rn[0] + rn[1]) + (rn[2] + rn[3])) + ((rn[4] + rn[5]) + (rn[6] + rn[7]));
    float am = ((rm[0] + rm[1]) + (rm[2] + rm[3])) + ((rm[4] + rm[5]) + (rm[6] + rm[7]));
    atomicAdd(&acc[0], an);
    atomicAdd(&acc[1], am);
  }
}

__global__ void finalize_k(const float* __restrict__ acc, float* __restrict__ out) {
  const float inv = 1.0f / (float)NCOL;
  float loss = acc[0] * inv;                     // mean NLL
  float mm   = acc[1] * inv;                     // mean mask
  out[0] = loss + loss * mm;                     // loss * (1 + mean(mask))
}

extern "C" void kernel_launch(void* const* d_in, const int* in_sizes, int n_in,
                              void* d_out, int out_size, void* d_ws, size_t ws_size,
                              hipStream_t stream) {
  (void)in_sizes; (void)n_in; (void)out_size; (void)ws_size;
  const float* X      = (const float*)d_in[0];
  const int*   target = (const int*)d_in[1];
  float* acc = (float*)d_ws;        // [0]=nll sum, [1]=mask sum
  float* lse = acc + 16;            // B*V = 16576 floats of scratch
  float* out = (float*)d_out;

  hipLaunchKernelGGL(zero_acc_k, dim3(1), dim3(1), 0, stream, acc);
  hipLaunchKernelGGL(pass1_lse_k, dim3(BB * VV), dim3(256), 0, stream, X, lse);
  hipLaunchKernelGGL(pass2_ce_k, dim3(BB * (SS / 256)), dim3(256), 0, stream,
                     X, target, lse, acc);
  hipLaunchKernelGGL(finalize_k, dim3(1), dim3(1), 0, stream, acc, out);
}